// AssignTarget_8143257993833
// MI455X (gfx1250) — compile-verified
//
#include <hip/hip_runtime.h>
#include <hip/hip_bf16.h>

typedef __attribute__((ext_vector_type(2))) float v2f;
typedef __attribute__((ext_vector_type(8))) float v8f;

#define ROWS 512
#define COLS 512
#define NPIX (ROWS * COLS)
#define NBOX 64
#define NB 2

__device__ __forceinline__ int imin(int a, int b) { return a < b ? a : b; }

// ---------------------------------------------------------------------------
// init: zero per-box counters in workspace (ws is poisoned by harness)
// ---------------------------------------------------------------------------
__global__ void at_init_counts(float* __restrict__ counts) {
    int t = threadIdx.x;
    if (t < NB * NBOX) counts[t] = 0.0f;
}

// ---------------------------------------------------------------------------
// Pass A: WMMA-based pixel->box-frame transform + first-hit index + counts.
// One wave32 handles a tile of 16 consecutive pixels.
//   A  [16x4] row m: [px_m, py_m, 1, 0]
//   Blx [4x16] col n: [c_n, s_n, tx_n, 0]   (tx = -c*bx - s*by)
//   Bly [4x16] col n: [-s_n, c_n, ty_n, 0]  (ty =  s*bx - c*by)
//   D = A x B : D[m][n] = local coordinate of pixel m in box n's frame.
// B fragments are prebuilt in LDS in the exact per-lane WMMA register layout
// (f32 4x16 B: lanes 0-15 carry rows K0,K1 of their column; lanes 16-31 carry
// rows K2,K3), so the hot loop is branchless: one ds_load_b64 per fragment.
// ---------------------------------------------------------------------------
__global__ __launch_bounds__(256)
void at_pass_a(const float* __restrict__ gt, const float* __restrict__ ext,
               int* __restrict__ idxmap, float* __restrict__ counts) {
    __shared__ v2f  sBlx[4 * 32];   // [group][lane] B fragment for lx
    __shared__ v2f  sBly[4 * 32];   // [group][lane] B fragment for ly
    __shared__ float shl[NBOX], shw[NBOX];

    const int tid  = threadIdx.x;
    const int wave = tid >> 5;
    const int lane = tid & 31;
    const int half = lane >> 4;      // 0: lanes 0-15, 1: lanes 16-31
    const int l16  = lane & 15;

    // 2048 blocks per batch, 8 tiles (waves) per block -> 16384 tiles per batch
    const int batch       = blockIdx.x >> 11;
    const int tileInBatch = ((blockIdx.x & 2047) << 3) + wave;

    // --- build B fragments in WMMA register layout (branch only in setup) ---
    if (tid < 128) {
        const int fl   = tid & 31;       // fragment lane
        const int fg   = tid >> 5;       // group 0..3
        const int fh   = fl >> 4;
        const int fbox = (fg << 4) + (fl & 15);
        const float* bp = gt + (size_t)(batch * NBOX + fbox) * 10;
        const float bx = bp[0], by = bp[1], yaw = bp[6];
        const float c = cosf(yaw), s = sinf(yaw);
        v2f blx, bly;
        if (fh == 0) { blx.x =  c;             blx.y = s;    // rows K0,K1
                       bly.x = -s;             bly.y = c;    }
        else         { blx.x = -c * bx - s * by; blx.y = 0.0f; // rows K2,K3
                       bly.x =  s * bx - c * by; bly.y = 0.0f; }
        sBlx[tid] = blx;
        sBly[tid] = bly;
    }
    if (tid < NBOX) {
        const float* bp = gt + (size_t)(batch * NBOX + tid) * 10;
        const float dl = bp[3], dw = bp[4];
        shl[tid] = (dl > 0.0f) ? dl * 0.5f : -1.0f;  // fold dl>0 into half-extent
        shw[tid] = (dw > 0.0f) ? dw * 0.5f : -1.0f;
    }
    __syncthreads();

    const float x0 = ext[0], x1 = ext[1], y0 = ext[2], y1 = ext[3];
    const float sx = (x1 - x0) * (1.0f / COLS);
    const float sy = (y1 - y0) * (1.0f / ROWS);

    const int pbase = tileInBatch * 16;

    // A fragment (f32 16x4 layout): lanes 0-15 -> {K0=px, K1=py} of pixel M=l16,
    // lanes 16-31 -> {K2=1, K3=0}.  Branchless selects.
    v2f afrag;
    {
        int p = pbase + l16;
        int r = p >> 9;           // / COLS
        int c = p & (COLS - 1);
        float px = x0 + ((float)c + 0.5f) * sx;
        float py = y0 + ((float)r + 0.5f) * sy;
        afrag.x = half ? 1.0f : px;
        afrag.y = half ? 0.0f : py;
    }

    int minIdx[8];
#pragma unroll
    for (int v = 0; v < 8; ++v) minIdx[v] = NBOX;

#pragma unroll
    for (int g = 0; g < 4; ++g) {
        const v2f blx = sBlx[(g << 5) + lane];   // one ds_load_b64 each, uniform
        const v2f bly = sBly[(g << 5) + lane];

        v8f zero = {};
        v8f lx = __builtin_amdgcn_wmma_f32_16x16x4_f32(
            false, afrag, false, blx, (short)0, zero, false, false);
        v8f ly = __builtin_amdgcn_wmma_f32_16x16x4_f32(
            false, afrag, false, bly, (short)0, zero, false, false);

        const int box  = (g << 4) + l16;         // this lane's D-column
        const float hl = shl[box];
        const float hw = shw[box];
#pragma unroll
        for (int v = 0; v < 8; ++v) {
            // D layout: this lane's entry v is (pixel m = v + 8*half, box n = box)
            bool inside = (__builtin_fabsf(lx[v]) <= hl) &
                          (__builtin_fabsf(ly[v]) <= hw);
            minIdx[v] = imin(minIdx[v], inside ? box : NBOX);
        }
    }

    // Reduce min box index across the 16 lanes of each half (per-pixel).
#pragma unroll
    for (int v = 0; v < 8; ++v) {
        int m = minIdx[v];
        m = imin(m, __shfl_xor(m, 1, 32));
        m = imin(m, __shfl_xor(m, 2, 32));
        m = imin(m, __shfl_xor(m, 4, 32));
        m = imin(m, __shfl_xor(m, 8, 32));
        minIdx[v] = m;
    }

    if (l16 == 0) {
#pragma unroll
        for (int v = 0; v < 8; ++v) {
            int p = pbase + v + (half << 3);
            int m = minIdx[v];
            idxmap[(size_t)batch * NPIX + p] = (m < NBOX) ? m : -1;
            if (m < NBOX) atomicAdd(&counts[batch * NBOX + m], 1.0f);
        }
    }
}

// ---------------------------------------------------------------------------
// Pass B: stream the 12 target maps + category (bandwidth-bound)
// ---------------------------------------------------------------------------
__global__ __launch_bounds__(256)
void at_pass_b(const float* __restrict__ gt, const float* __restrict__ ext,
               const int* __restrict__ idxmap, const float* __restrict__ counts,
               float* __restrict__ maps, int* __restrict__ category) {
    int p = blockIdx.x * blockDim.x + threadIdx.x;
    if (p >= NB * NPIX) return;

    const int batch = p / NPIX;
    const int pix   = p - batch * NPIX;
    const int r     = pix >> 9;
    const int c     = pix & (COLS - 1);

    const float x0 = ext[0], x1 = ext[1], y0 = ext[2], y1 = ext[3];
    const float px = x0 + ((float)c + 0.5f) * (x1 - x0) * (1.0f / COLS);
    const float py = y0 + ((float)r + 0.5f) * (y1 - y0) * (1.0f / ROWS);

    const int idx    = idxmap[p];
    const float mask = (idx >= 0) ? 1.0f : 0.0f;
    const int sel    = (idx >= 0) ? idx : 0;

    const float* bp = gt + (size_t)(batch * NBOX + sel) * 10;
    const float bx = bp[0], by = bp[1], bz = bp[2];
    const float dl = bp[3], dw = bp[4], dh = bp[5];
    const float yaw = bp[6], vx = bp[7], vy = bp[8], cls = bp[9];

    const float cnt = counts[batch * NBOX + sel];
    const float ow  = mask / fmaxf(cnt, 1.0f);

    float ysin, ycos;
    __sincosf(yaw, &ysin, &ycos);

    const size_t S  = NPIX;
    float* mb = maps + (size_t)batch * 12 * NPIX + pix;
    mb[0 * S]  = mask;
    mb[1 * S]  = (bx - px) * mask;
    mb[2 * S]  = (by - py) * mask;
    mb[3 * S]  = ow;
    mb[4 * S]  = vx * mask;
    mb[5 * S]  = vy * mask;
    mb[6 * S]  = dl * mask;
    mb[7 * S]  = dw * mask;
    mb[8 * S]  = dh * mask;
    mb[9 * S]  = ysin * mask;
    mb[10 * S] = ycos * mask;
    mb[11 * S] = bz * mask;

    category[p] = (idx >= 0) ? (int)cls : -1;
}

// ---------------------------------------------------------------------------
// Export counts [B, NBOX] to output tail
// ---------------------------------------------------------------------------
__global__ void at_copy_counts(const float* __restrict__ counts,
                               float* __restrict__ out) {
    int t = threadIdx.x;
    if (t < NB * NBOX) out[t] = counts[t];
}

extern "C" void kernel_launch(void* const* d_in, const int* in_sizes, int n_in,
                              void* d_out, int out_size, void* d_ws, size_t ws_size,
                              hipStream_t stream) {
    (void)in_sizes; (void)n_in; (void)out_size; (void)ws_size;

    const float* gt  = (const float*)d_in[0];   // [B, NBOX, 10]
    const float* ext = (const float*)d_in[1];   // [3, 2]

    float* out        = (float*)d_out;
    float* maps       = out;                                          // [B,12,H,W]
    int*   category   = (int*)(out + (size_t)NB * 12 * NPIX);         // [B,H,W] int32 bits
    float* counts_out = out + (size_t)NB * 12 * NPIX + (size_t)NB * NPIX;  // [B,NBOX]

    int*   idxmap = (int*)d_ws;                                        // [B,H,W]
    float* counts = (float*)((char*)d_ws + sizeof(int) * (size_t)NB * NPIX);

    at_init_counts<<<1, 128, 0, stream>>>(counts);

    // 16384 tiles/batch, 8 waves (tiles) per 256-thread block -> 2048 blocks/batch
    at_pass_a<<<NB * 2048, 256, 0, stream>>>(gt, ext, idxmap, counts);

    at_pass_b<<<(NB * NPIX + 255) / 256, 256, 0, stream>>>(
        gt, ext, idxmap, counts, maps, category);

    at_copy_counts<<<1, 128, 0, stream>>>(counts, counts_out);
}